// CompGraphConvLayer_48395691491487
// MI455X (gfx1250) — compile-verified
//
#include <hip/hip_runtime.h>
#include <hip/hip_bf16.h>

typedef __attribute__((ext_vector_type(2))) float v2f;
typedef __attribute__((ext_vector_type(8))) float v8f;

#define DIN 256
#define DOUT 256

// ---------------------------------------------------------------------------
// Scatter: for each edge, S[dst] += x[src] (256 floats), deg[dst] += 1.
// 64 threads per edge, float4 per thread. Both relations in one grid.
// ---------------------------------------------------------------------------
__global__ __launch_bounds__(256) void scatter_kernel(
    const int* __restrict__ src0, const int* __restrict__ dst0,
    const int* __restrict__ src1, const int* __restrict__ dst1,
    const float* __restrict__ x,
    float* __restrict__ S0, float* __restrict__ S1,
    float* __restrict__ deg0, float* __restrict__ deg1, int E)
{
    int g = blockIdx.x * 4 + (threadIdx.x >> 6);   // global edge slot
    int t = threadIdx.x & 63;                      // lane within edge group
    if (g >= 2 * E) return;

    const int* src; const int* dst; float* S; float* deg; int e;
    if (g < E) { e = g;     src = src0; dst = dst0; S = S0; deg = deg0; }
    else       { e = g - E; src = src1; dst = dst1; S = S1; deg = deg1; }

    int s = src[e];
    int d = dst[e];
    const float4 xv = ((const float4*)(x + (size_t)s * DIN))[t];
    float* Sp = S + (size_t)d * DIN + t * 4;
    atomicAdd(Sp + 0, xv.x);
    atomicAdd(Sp + 1, xv.y);
    atomicAdd(Sp + 2, xv.z);
    atomicAdd(Sp + 3, xv.w);
    if (t == 0) atomicAdd(deg + d, 1.0f);
}

// ---------------------------------------------------------------------------
// Constant vectors: v1 = r1@WO, v2 = r2@WI, cvec = WSb+WOb+WIb - r3@WS.
// ---------------------------------------------------------------------------
__global__ __launch_bounds__(256) void consts_kernel(
    const float* __restrict__ r_feats,
    const float* __restrict__ WS, const float* __restrict__ WO, const float* __restrict__ WI,
    const float* __restrict__ WSb, const float* __restrict__ WOb, const float* __restrict__ WIb,
    float* __restrict__ v1, float* __restrict__ v2, float* __restrict__ cvec)
{
    int j = threadIdx.x;
    const float* r1 = r_feats + 1 * DIN;
    const float* r2 = r_feats + 2 * DIN;
    const float* r3 = r_feats + 3 * DIN;   // last relation embedding
    float a1 = 0.f, a2 = 0.f, a3 = 0.f;
    for (int k = 0; k < DIN; ++k) {
        a1 = fmaf(r1[k], WO[k * DOUT + j], a1);
        a2 = fmaf(r2[k], WI[k * DOUT + j], a2);
        a3 = fmaf(r3[k], WS[k * DOUT + j], a3);
    }
    v1[j] = a1;
    v2[j] = a2;
    cvec[j] = WSb[j] + WOb[j] + WIb[j] - a3;
}

// ---------------------------------------------------------------------------
// Fused triple GEMM + epilogue via V_WMMA_F32_16X16X4_F32.
// One wave computes a 64x16 tile of n_out (4 M-subtiles, register-blocked):
//   n_out = X@WS + diag(inv0)(S0@WO) + diag(inv1)(S1@WI)
//           + cvec - has0*v1 - has1*v2
// B fragments are loaded once per k-step and reused across the 4 M-subtiles.
// ---------------------------------------------------------------------------
__global__ __launch_bounds__(256) void gemm_wmma_kernel(
    const float* __restrict__ x,
    const float* __restrict__ S0, const float* __restrict__ S1,
    const float* __restrict__ deg0, const float* __restrict__ deg1,
    const float* __restrict__ WS, const float* __restrict__ WO, const float* __restrict__ WI,
    const float* __restrict__ v1, const float* __restrict__ v2, const float* __restrict__ cvec,
    float* __restrict__ out, int N)
{
    const int wave = threadIdx.x >> 5;            // 0..7
    const int lane = threadIdx.x & 31;
    const int job  = blockIdx.x * 8 + wave;
    const int nt   = job & 15;                    // N-tile (DOUT/16 == 16)
    const int mg   = job >> 4;                    // group of 4 M-tiles
    const int m0   = mg * 64;
    if (m0 >= N) return;                          // wave-uniform

    // f32 WMMA A(16x4): lane l -> row l&15, K-pair 2*(l>>4); B(4x16) symmetric.
    const int lrow = lane & 15;
    const int kh   = (lane >> 4) * 2;             // 0 or 2
    const int colB = nt * 16 + lrow;

    // Clamped subtile starts (wave-uniform per i; N is a multiple of 16, so
    // each subtile is either fully valid or fully clamped-duplicate).
    int tstart[4];
#pragma unroll
    for (int i = 0; i < 4; ++i) {
        int t = m0 + 16 * i;
        tstart[i] = (t > N - 16) ? (N - 16) : t;
    }

    const float* bS = WS + colB;
    const float* bO = WO + colB;
    const float* bI = WI + colB;

    v8f accS[4] = {{0}, {0}, {0}, {0}};
    v8f accO[4] = {{0}, {0}, {0}, {0}};
    v8f accI[4] = {{0}, {0}, {0}, {0}};

#pragma unroll 2
    for (int k = 0; k < DIN; k += 4) {
        const int ka = k + kh;
        v2f fS = { bS[(size_t)ka * DOUT], bS[(size_t)(ka + 1) * DOUT] };
        v2f fO = { bO[(size_t)ka * DOUT], bO[(size_t)(ka + 1) * DOUT] };
        v2f fI = { bI[(size_t)ka * DOUT], bI[(size_t)(ka + 1) * DOUT] };
#pragma unroll
        for (int i = 0; i < 4; ++i) {
            const size_t rbase = (size_t)(tstart[i] + lrow) * DIN + ka;
            v2f aS = { x[rbase],  x[rbase + 1]  };
            v2f aO = { S0[rbase], S0[rbase + 1] };
            v2f aI = { S1[rbase], S1[rbase + 1] };
            accS[i] = __builtin_amdgcn_wmma_f32_16x16x4_f32(
                false, aS, false, fS, (short)0, accS[i], false, false);
            accO[i] = __builtin_amdgcn_wmma_f32_16x16x4_f32(
                false, aO, false, fO, (short)0, accO[i], false, false);
            accI[i] = __builtin_amdgcn_wmma_f32_16x16x4_f32(
                false, aI, false, fI, (short)0, accI[i], false, false);
        }
    }

    // Epilogue. C/D layout: VGPR r, lane l -> row = r + 8*(l>>4), col = l&15.
    const float cj  = cvec[colB];
    const float v1j = v1[colB];
    const float v2j = v2[colB];
#pragma unroll
    for (int i = 0; i < 4; ++i) {
        const int t0 = m0 + 16 * i;
        if (t0 >= N) break;                       // wave-uniform
        const int rb = t0 + ((lane >> 4) << 3);
#pragma unroll
        for (int r = 0; r < 8; ++r) {
            const int row = rb + r;
            const float d0 = deg0[row];
            const float d1 = deg1[row];
            const float i0 = 1.0f / fmaxf(d0, 1.0f);
            const float i1 = 1.0f / fmaxf(d1, 1.0f);
            const float h0 = (d0 > 0.f) ? 1.f : 0.f;
            const float h1 = (d1 > 0.f) ? 1.f : 0.f;
            out[(size_t)row * DOUT + colB] =
                accS[i][r] + i0 * accO[i][r] + i1 * accI[i][r]
                + cj - h0 * v1j - h1 * v2j;
        }
    }
}

// ---------------------------------------------------------------------------
// r_out = r_feats @ WR + WRb   (4 x 256) — tiny, scalar dot per output.
// ---------------------------------------------------------------------------
__global__ __launch_bounds__(256) void rout_kernel(
    const float* __restrict__ r_feats, const float* __restrict__ WR,
    const float* __restrict__ WRb, float* __restrict__ r_out)
{
    int j = threadIdx.x;
    int i = blockIdx.x;     // 0..3
    float a = WRb[j];
    for (int k = 0; k < DIN; ++k)
        a = fmaf(r_feats[i * DIN + k], WR[k * DOUT + j], a);
    r_out[i * DOUT + j] = a;
}

// ---------------------------------------------------------------------------
extern "C" void kernel_launch(void* const* d_in, const int* in_sizes, int n_in,
                              void* d_out, int out_size, void* d_ws, size_t ws_size,
                              hipStream_t stream)
{
    const float* x       = (const float*)d_in[0];
    const float* r_feats = (const float*)d_in[1];
    const int*   src0    = (const int*)d_in[2];
    const int*   dst0    = (const int*)d_in[3];
    const int*   src1    = (const int*)d_in[4];
    const int*   dst1    = (const int*)d_in[5];
    const float* WO_w    = (const float*)d_in[6];
    const float* WO_b    = (const float*)d_in[7];
    const float* WI_w    = (const float*)d_in[8];
    const float* WI_b    = (const float*)d_in[9];
    const float* WS_w    = (const float*)d_in[10];
    const float* WS_b    = (const float*)d_in[11];
    const float* WR_w    = (const float*)d_in[12];
    const float* WR_b    = (const float*)d_in[13];

    const int N = in_sizes[0] / DIN;   // 50000
    const int E = in_sizes[2];         // 200000

    float* ws   = (float*)d_ws;
    float* S0   = ws;
    float* S1   = S0 + (size_t)N * DIN;
    float* deg0 = S1 + (size_t)N * DIN;
    float* deg1 = deg0 + N;
    float* v1   = deg1 + N;
    float* v2   = v1 + DOUT;
    float* cvec = v2 + DOUT;

    float* n_out = (float*)d_out;
    float* r_out = n_out + (size_t)N * DOUT;

    // Zero accumulators + degrees (graph-capture-safe async memset).
    hipMemsetAsync(S0, 0, ((size_t)2 * N * DIN + 2 * (size_t)N) * sizeof(float), stream);

    // Edge scatter for both relations.
    scatter_kernel<<<(2 * E + 3) / 4, 256, 0, stream>>>(
        src0, dst0, src1, dst1, x, S0, S1, deg0, deg1, E);

    // Constant epilogue vectors.
    consts_kernel<<<1, 256, 0, stream>>>(
        r_feats, WS_w, WO_w, WI_w, WS_b, WO_b, WI_b, v1, v2, cvec);

    // Fused triple GEMM with f32 WMMA, 64x16 per wave.
    const int mgroups = (N / 16 + 3) / 4;          // 782
    const int jobs    = mgroups * (DOUT / 16);     // 12512
    gemm_wmma_kernel<<<(jobs + 7) / 8, 256, 0, stream>>>(
        x, S0, S1, deg0, deg1, WS_w, WO_w, WI_w, v1, v2, cvec, n_out, N);

    // Relation embedding update.
    rout_kernel<<<4, 256, 0, stream>>>(r_feats, WR_w, WR_b, r_out);
}